// LocalBlock_71055938945436
// MI455X (gfx1250) — compile-verified
//
#include <hip/hip_runtime.h>
#include <hip/hip_bf16.h>
#include <math.h>

#define NN 20000
#define NE 320000

typedef __attribute__((ext_vector_type(2))) float v2f;
typedef __attribute__((ext_vector_type(8))) float v8f;

// ---------------------------------------------------------------------------
// fp32 WMMA 16x16x4 accumulate helper.
// A fragment: lane holds M = lane&15, K = {kg, kg+1}, kg = (lane>>4)*2
// B fragment: lane holds N = lane&15, K = {kg, kg+1}
// D fragment: reg i -> lanes 0-15: (M=i, N=lane); lanes 16-31: (M=i+8, N=lane-16)
// ---------------------------------------------------------------------------
__device__ __forceinline__ v8f wmma_rowcol_f32(const float* __restrict__ Arow,
                                               const float* __restrict__ Brow,
                                               int Ci, int kg, v8f acc) {
  for (int k = 0; k < Ci; k += 4) {
    v2f a, b;
    a[0] = Arow[k + kg];
    a[1] = Arow[k + kg + 1];
    b[0] = Brow[k + kg];
    b[1] = Brow[k + kg + 1];
    acc = __builtin_amdgcn_wmma_f32_16x16x4_f32(false, a, false, b, (short)0, acc,
                                                false, false);
  }
  return acc;
}

__device__ __forceinline__ void atomicMaxF(float* addr, float v) {
  if (v >= 0.0f)
    atomicMax((int*)addr, __float_as_int(v));
  else
    atomicMin((unsigned int*)addr, __float_as_uint(v));
}

// ---------------------------------------------------------------------------
// Generic node GEMM: Y[n,Co] = act(X[n,Ci] @ W[Co,Ci]^T + b). act: 0=id 1=sigmoid
// One wave per 16x16 output tile.
// ---------------------------------------------------------------------------
__global__ void node_gemm_kernel(const float* __restrict__ X, const float* __restrict__ W,
                                 const float* __restrict__ bias, float* __restrict__ Y,
                                 int n, int Ci, int Co, int act) {
  int lane = threadIdx.x & 31;
  int wave = blockIdx.x * (blockDim.x >> 5) + (threadIdx.x >> 5);
  int colTiles = Co >> 4;
  int total = (n >> 4) * colTiles;
  if (wave >= total) return;
  int rt = wave / colTiles, ct = wave - rt * colTiles;
  int r0 = rt << 4, c0 = ct << 4;
  int lm = lane & 15;
  int kg = (lane >> 4) << 1;
  const float* Arow = X + (size_t)(r0 + lm) * Ci;
  const float* Brow = W + (size_t)(c0 + lm) * Ci;
  v8f acc = {};
  acc = wmma_rowcol_f32(Arow, Brow, Ci, kg, acc);
  int col = c0 + lm;
  int rbase = r0 + ((lane >> 4) << 3);
  float bv = bias[col];
#pragma unroll
  for (int i = 0; i < 8; ++i) {
    float v = acc[i] + bv;
    if (act == 1) v = 1.0f / (1.0f + __expf(-v));
    Y[(size_t)(rbase + i) * Co + col] = v;
  }
}

// ---------------------------------------------------------------------------
// GINE fused message: AGG[dst] += relu(X[src] + EA @ We^T + be), Ci(edge)=64
// ---------------------------------------------------------------------------
__global__ void gine_msg_kernel(const float* __restrict__ EA, const float* __restrict__ We,
                                const float* __restrict__ be, const float* __restrict__ X,
                                const int* __restrict__ src, const int* __restrict__ dst,
                                float* __restrict__ AGG, int Co) {
  const int Ci = 64;
  int lane = threadIdx.x & 31;
  int wave = blockIdx.x * (blockDim.x >> 5) + (threadIdx.x >> 5);
  int colTiles = Co >> 4;
  int total = (NE >> 4) * colTiles;
  if (wave >= total) return;
  int rt = wave / colTiles, ct = wave - rt * colTiles;
  int r0 = rt << 4, c0 = ct << 4;
  int lm = lane & 15;
  int kg = (lane >> 4) << 1;
  const float* Arow = EA + (size_t)(r0 + lm) * Ci;
  const float* Brow = We + (size_t)(c0 + lm) * Ci;
  v8f acc = {};
  acc = wmma_rowcol_f32(Arow, Brow, Ci, kg, acc);
  int col = c0 + lm;
  int rbase = r0 + ((lane >> 4) << 3);
  float bv = be[col];
#pragma unroll
  for (int i = 0; i < 8; ++i) {
    int e = rbase + i;
    int s = src[e], d = dst[e];
    float m = acc[i] + bv + X[(size_t)s * Co + col];
    m = fmaxf(m, 0.0f);
    atomicAdd(&AGG[(size_t)d * Co + col], m);
  }
}

// H = (1+eps)*X + AGG, elementwise
__global__ void gine_combine_kernel(const float* __restrict__ X, const float* __restrict__ AGG,
                                    const float* __restrict__ epsPtr, float* __restrict__ H,
                                    int total) {
  int idx = blockIdx.x * blockDim.x + threadIdx.x;
  if (idx >= total) return;
  float e = 1.0f + epsPtr[0];
  H[idx] = e * X[idx] + AGG[idx];
}

// ---------------------------------------------------------------------------
// GATv2 logit: LOGIT[e] = sum_c att[c]*leaky_relu(xl[src,c]+xr[dst,c]+(ea@We^T)[e,c])
// One wave per 16-edge tile; loops the 8 column tiles (Co=128), reduces across lanes.
// ---------------------------------------------------------------------------
__global__ void gat_logit_kernel(const float* __restrict__ EA, const float* __restrict__ We,
                                 const float* __restrict__ XL, const float* __restrict__ XR,
                                 const float* __restrict__ att, const int* __restrict__ src,
                                 const int* __restrict__ dst, float* __restrict__ LOGIT) {
  const int Ci = 64, Co = 128;
  int lane = threadIdx.x & 31;
  int wave = blockIdx.x * (blockDim.x >> 5) + (threadIdx.x >> 5);
  if (wave >= (NE >> 4)) return;
  int r0 = wave << 4;
  int lm = lane & 15;
  int kg = (lane >> 4) << 1;
  int rbase = r0 + ((lane >> 4) << 3);
  const float* Arow = EA + (size_t)(r0 + lm) * Ci;
  float rowsum[8];
#pragma unroll
  for (int i = 0; i < 8; ++i) rowsum[i] = 0.0f;
  for (int ct = 0; ct < Co / 16; ++ct) {
    int col = (ct << 4) + lm;
    const float* Brow = We + (size_t)col * Ci;
    v8f acc = {};
    acc = wmma_rowcol_f32(Arow, Brow, Ci, kg, acc);
    float av = att[col];
#pragma unroll
    for (int i = 0; i < 8; ++i) {
      int e = rbase + i;
      int s = src[e], d = dst[e];
      float z = acc[i] + XL[(size_t)s * Co + col] + XR[(size_t)d * Co + col];
      z = (z > 0.0f) ? z : 0.2f * z;
      rowsum[i] += z * av;
    }
  }
#pragma unroll
  for (int off = 8; off > 0; off >>= 1)
#pragma unroll
    for (int i = 0; i < 8; ++i) rowsum[i] += __shfl_xor(rowsum[i], off, 16);
  if (lm == 0) {
#pragma unroll
    for (int i = 0; i < 8; ++i) LOGIT[rbase + i] = rowsum[i];
  }
}

// GATv2 output accumulation: OUT[dst] += alpha * XL[src]   (one wave per edge)
__global__ void gat_accum_kernel(const float* __restrict__ XL, const int* __restrict__ src,
                                 const int* __restrict__ dst, const float* __restrict__ EX,
                                 const float* __restrict__ SS, float* __restrict__ OUT) {
  const int Co = 128;
  int e = blockIdx.x * (blockDim.x >> 5) + (threadIdx.x >> 5);
  if (e >= NE) return;
  int lane = threadIdx.x & 31;
  int s = src[e], d = dst[e];
  float a = EX[e] / (SS[d] + 1e-16f);
  for (int c = lane; c < Co; c += 32)
    atomicAdd(&OUT[(size_t)d * Co + c], XL[(size_t)s * Co + c] * a);
}

// ---------------------------------------------------------------------------
// TransformerConv logits: per wave (16-edge tile, head h of 4; D=64, Cfull=256)
// kj = (ea@We^T)+be+K[src]; logit = sum_c q[dst,c]*kj[c] * 0.125
// ---------------------------------------------------------------------------
__global__ void trf_logit_kernel(const float* __restrict__ EA, const float* __restrict__ We,
                                 const float* __restrict__ be, const float* __restrict__ Q,
                                 const float* __restrict__ K, const int* __restrict__ src,
                                 const int* __restrict__ dst, float* __restrict__ LOGIT) {
  const int Ci = 64, Cf = 256;
  int lane = threadIdx.x & 31;
  int wave = blockIdx.x * (blockDim.x >> 5) + (threadIdx.x >> 5);
  int total = (NE >> 4) * 4;
  if (wave >= total) return;
  int rt = wave >> 2, h = wave & 3;
  int r0 = rt << 4;
  int lm = lane & 15;
  int kg = (lane >> 4) << 1;
  int rbase = r0 + ((lane >> 4) << 3);
  const float* Arow = EA + (size_t)(r0 + lm) * Ci;
  float rowsum[8];
#pragma unroll
  for (int i = 0; i < 8; ++i) rowsum[i] = 0.0f;
  for (int ct = 0; ct < 4; ++ct) {
    int col = (h << 6) + (ct << 4) + lm;
    const float* Brow = We + (size_t)col * Ci;
    v8f acc = {};
    acc = wmma_rowcol_f32(Arow, Brow, Ci, kg, acc);
    float bv = be[col];
#pragma unroll
    for (int i = 0; i < 8; ++i) {
      int e = rbase + i;
      int s = src[e], d = dst[e];
      float kj = acc[i] + bv + K[(size_t)s * Cf + col];
      rowsum[i] += kj * Q[(size_t)d * Cf + col];
    }
  }
#pragma unroll
  for (int off = 8; off > 0; off >>= 1)
#pragma unroll
    for (int i = 0; i < 8; ++i) rowsum[i] += __shfl_xor(rowsum[i], off, 16);
  if (lm == 0) {
#pragma unroll
    for (int i = 0; i < 8; ++i) LOGIT[(size_t)(rbase + i) * 4 + h] = rowsum[i] * 0.125f;
  }
}

// Transformer message: OUT[dst] += alpha[e,h] * ((ea@We^T)+be+V[src])
__global__ void trf_msg_kernel(const float* __restrict__ EA, const float* __restrict__ We,
                               const float* __restrict__ be, const float* __restrict__ V,
                               const int* __restrict__ src, const int* __restrict__ dst,
                               const float* __restrict__ EX, const float* __restrict__ SS,
                               float* __restrict__ OUT) {
  const int Ci = 64, Cf = 256;
  int lane = threadIdx.x & 31;
  int wave = blockIdx.x * (blockDim.x >> 5) + (threadIdx.x >> 5);
  int total = (NE >> 4) * (Cf >> 4);
  if (wave >= total) return;
  int rt = wave >> 4, ct = wave & 15;
  int r0 = rt << 4, c0 = ct << 4;
  int lm = lane & 15;
  int kg = (lane >> 4) << 1;
  int rbase = r0 + ((lane >> 4) << 3);
  int col = c0 + lm;
  int h = col >> 6;
  const float* Arow = EA + (size_t)(r0 + lm) * Ci;
  const float* Brow = We + (size_t)col * Ci;
  v8f acc = {};
  acc = wmma_rowcol_f32(Arow, Brow, Ci, kg, acc);
  float bv = be[col];
#pragma unroll
  for (int i = 0; i < 8; ++i) {
    int e = rbase + i;
    int s = src[e], d = dst[e];
    float alpha = EX[(size_t)e * 4 + h] / (SS[(size_t)d * 4 + h] + 1e-16f);
    float val = acc[i] + bv + V[(size_t)s * Cf + col];
    atomicAdd(&OUT[(size_t)d * Cf + col], val * alpha);
  }
}

// ---------------------------------------------------------------------------
// Segment softmax helpers (H heads)
// ---------------------------------------------------------------------------
__global__ void seg_max_kernel(const float* __restrict__ LOGIT, const int* __restrict__ dst,
                               float* __restrict__ SM, int H) {
  int idx = blockIdx.x * blockDim.x + threadIdx.x;
  if (idx >= NE * H) return;
  int e = idx / H, h = idx - e * H;
  atomicMaxF(&SM[(size_t)dst[e] * H + h], LOGIT[idx]);
}

__global__ void seg_exp_kernel(const float* __restrict__ LOGIT, const int* __restrict__ dst,
                               const float* __restrict__ SM, float* __restrict__ EX,
                               float* __restrict__ SS, int H) {
  int idx = blockIdx.x * blockDim.x + threadIdx.x;
  if (idx >= NE * H) return;
  int e = idx / H, h = idx - e * H;
  float v = __expf(LOGIT[idx] - SM[(size_t)dst[e] * H + h]);
  EX[idx] = v;
  atomicAdd(&SS[(size_t)dst[e] * H + h], v);
}

__global__ void fill_kernel(float* __restrict__ p, float v, int total) {
  int idx = blockIdx.x * blockDim.x + threadIdx.x;
  if (idx < total) p[idx] = v;
}

__global__ void bias_init_kernel(float* __restrict__ Y, const float* __restrict__ b,
                                 int total, int C) {
  int idx = blockIdx.x * blockDim.x + threadIdx.x;
  if (idx < total) Y[idx] = b[idx % C];
}

// ---------------------------------------------------------------------------
extern "C" void kernel_launch(void* const* d_in, const int* in_sizes, int n_in,
                              void* d_out, int out_size, void* d_ws, size_t ws_size,
                              hipStream_t stream) {
  const float* x  = (const float*)d_in[0];
  const int*   ei = (const int*)d_in[1];
  const float* ea = (const float*)d_in[2];
  const int* src = ei;
  const int* dst = ei + NE;

  // c1 GINE
  const float *c1We = (const float*)d_in[3], *c1be = (const float*)d_in[4];
  const float *c1eps = (const float*)d_in[5];
  const float *c1W1 = (const float*)d_in[6], *c1b1 = (const float*)d_in[7];
  const float *c1W2 = (const float*)d_in[8], *c1b2 = (const float*)d_in[9];
  // c2 GINE
  const float *c2We = (const float*)d_in[10], *c2be = (const float*)d_in[11];
  const float *c2eps = (const float*)d_in[12];
  const float *c2W1 = (const float*)d_in[13], *c2b1 = (const float*)d_in[14];
  const float *c2W2 = (const float*)d_in[15], *c2b2 = (const float*)d_in[16];
  // c3 GATv2
  const float *c3Wl = (const float*)d_in[17], *c3bl = (const float*)d_in[18];
  const float *c3Wr = (const float*)d_in[19], *c3br = (const float*)d_in[20];
  const float *c3We = (const float*)d_in[21], *c3att = (const float*)d_in[22];
  const float *c3bias = (const float*)d_in[23];
  // c4 Transformer
  const float *c4Wq = (const float*)d_in[24], *c4bq = (const float*)d_in[25];
  const float *c4Wk = (const float*)d_in[26], *c4bk = (const float*)d_in[27];
  const float *c4Wv = (const float*)d_in[28], *c4bv = (const float*)d_in[29];
  const float *c4We = (const float*)d_in[30], *c4be = (const float*)d_in[31];
  const float *c4Ws = (const float*)d_in[32], *c4bs = (const float*)d_in[33];
  // c5 Transformer
  const float *c5Wq = (const float*)d_in[34], *c5bq = (const float*)d_in[35];
  const float *c5Wk = (const float*)d_in[36], *c5bk = (const float*)d_in[37];
  const float *c5Wv = (const float*)d_in[38], *c5bv = (const float*)d_in[39];
  const float *c5We = (const float*)d_in[40], *c5be = (const float*)d_in[41];
  const float *c5Ws = (const float*)d_in[42], *c5bs = (const float*)d_in[43];
  // c6 GINE
  const float *c6We = (const float*)d_in[44], *c6be = (const float*)d_in[45];
  const float *c6eps = (const float*)d_in[46];
  const float *c6W1 = (const float*)d_in[47], *c6b1 = (const float*)d_in[48];
  const float *c6W2 = (const float*)d_in[49], *c6b2 = (const float*)d_in[50];

  // workspace layout
  size_t NB = (size_t)NN * 256;
  float* P0 = (float*)d_ws;
  float* P1 = P0 + NB;
  float* P2 = P1 + NB;
  float* P3 = P2 + NB;
  float* P4 = P3 + NB;
  float* EL = P4 + NB;               // [E,4] logits
  float* EX = EL + (size_t)NE * 4;   // [E,4] exp
  float* SM = EX + (size_t)NE * 4;   // [N,4] seg max
  float* SS = SM + (size_t)NN * 4;   // [N,4] seg sum

  const float NEG_INF = -3.402823466e38f;

  auto gemm = [&](const float* X, const float* W, const float* b, float* Y,
                  int n, int Ci, int Co, int act) {
    int waves = (n >> 4) * (Co >> 4);
    node_gemm_kernel<<<(waves + 7) / 8, 256, 0, stream>>>(X, W, b, Y, n, Ci, Co, act);
  };
  auto fill = [&](float* p, float v, int total) {
    fill_kernel<<<(total + 255) / 256, 256, 0, stream>>>(p, v, total);
  };

  // ---------------- c1: GINE 128 -> 64 ----------------
  fill(P0, 0.0f, NN * 128);
  {
    int waves = (NE >> 4) * (128 >> 4);
    gine_msg_kernel<<<(waves + 7) / 8, 256, 0, stream>>>(ea, c1We, c1be, x, src, dst, P0, 128);
  }
  gine_combine_kernel<<<(NN * 128 + 255) / 256, 256, 0, stream>>>(x, P0, c1eps, P1, NN * 128);
  gemm(P1, c1W1, c1b1, P2, NN, 128, 96, 1);
  gemm(P2, c1W2, c1b2, P3, NN, 96, 64, 0);

  // ---------------- c2: GINE 64 -> 64 ----------------
  fill(P0, 0.0f, NN * 64);
  {
    int waves = (NE >> 4) * (64 >> 4);
    gine_msg_kernel<<<(waves + 7) / 8, 256, 0, stream>>>(ea, c2We, c2be, P3, src, dst, P0, 64);
  }
  gine_combine_kernel<<<(NN * 64 + 255) / 256, 256, 0, stream>>>(P3, P0, c2eps, P1, NN * 64);
  gemm(P1, c2W1, c2b1, P2, NN, 64, 32, 1);
  gemm(P2, c2W2, c2b2, P4, NN, 32, 64, 0);

  // ---------------- c3: GATv2 64 -> 128 ----------------
  gemm(P4, c3Wl, c3bl, P0, NN, 64, 128, 0);   // xl
  gemm(P4, c3Wr, c3br, P1, NN, 64, 128, 0);   // xr
  {
    int waves = NE >> 4;
    gat_logit_kernel<<<(waves + 7) / 8, 256, 0, stream>>>(ea, c3We, P0, P1, c3att, src, dst, EL);
  }
  fill(SM, NEG_INF, NN);
  seg_max_kernel<<<(NE + 255) / 256, 256, 0, stream>>>(EL, dst, SM, 1);
  fill(SS, 0.0f, NN);
  seg_exp_kernel<<<(NE + 255) / 256, 256, 0, stream>>>(EL, dst, SM, EX, SS, 1);
  bias_init_kernel<<<(NN * 128 + 255) / 256, 256, 0, stream>>>(P2, c3bias, NN * 128, 128);
  gat_accum_kernel<<<(NE + 7) / 8, 256, 0, stream>>>(P0, src, dst, EX, SS, P2);

  // ---------------- c4: TransformerConv 128 -> 256 ----------------
  gemm(P2, c4Wq, c4bq, P0, NN, 128, 256, 0);  // q
  gemm(P2, c4Wk, c4bk, P1, NN, 128, 256, 0);  // k
  gemm(P2, c4Wv, c4bv, P3, NN, 128, 256, 0);  // v
  gemm(P2, c4Ws, c4bs, P4, NN, 128, 256, 0);  // skip -> out accumulator
  {
    int waves = (NE >> 4) * 4;
    trf_logit_kernel<<<(waves + 7) / 8, 256, 0, stream>>>(ea, c4We, c4be, P0, P1, src, dst, EL);
  }
  fill(SM, NEG_INF, NN * 4);
  seg_max_kernel<<<(NE * 4 + 255) / 256, 256, 0, stream>>>(EL, dst, SM, 4);
  fill(SS, 0.0f, NN * 4);
  seg_exp_kernel<<<(NE * 4 + 255) / 256, 256, 0, stream>>>(EL, dst, SM, EX, SS, 4);
  {
    int waves = (NE >> 4) * 16;
    trf_msg_kernel<<<(waves + 7) / 8, 256, 0, stream>>>(ea, c4We, c4be, P3, src, dst, EX, SS, P4);
  }

  // ---------------- c5: TransformerConv 256 -> 256 ----------------
  gemm(P4, c5Wq, c5bq, P0, NN, 256, 256, 0);
  gemm(P4, c5Wk, c5bk, P1, NN, 256, 256, 0);
  gemm(P4, c5Wv, c5bv, P3, NN, 256, 256, 0);
  gemm(P4, c5Ws, c5bs, P2, NN, 256, 256, 0);  // skip -> out accumulator
  {
    int waves = (NE >> 4) * 4;
    trf_logit_kernel<<<(waves + 7) / 8, 256, 0, stream>>>(ea, c5We, c5be, P0, P1, src, dst, EL);
  }
  fill(SM, NEG_INF, NN * 4);
  seg_max_kernel<<<(NE * 4 + 255) / 256, 256, 0, stream>>>(EL, dst, SM, 4);
  fill(SS, 0.0f, NN * 4);
  seg_exp_kernel<<<(NE * 4 + 255) / 256, 256, 0, stream>>>(EL, dst, SM, EX, SS, 4);
  {
    int waves = (NE >> 4) * 16;
    trf_msg_kernel<<<(waves + 7) / 8, 256, 0, stream>>>(ea, c5We, c5be, P3, src, dst, EX, SS, P2);
  }

  // ---------------- c6: GINE 256 -> 64 ----------------
  fill(P0, 0.0f, NN * 256);
  {
    int waves = (NE >> 4) * (256 >> 4);
    gine_msg_kernel<<<(waves + 7) / 8, 256, 0, stream>>>(ea, c6We, c6be, P2, src, dst, P0, 256);
  }
  gine_combine_kernel<<<(NN * 256 + 255) / 256, 256, 0, stream>>>(P2, P0, c6eps, P1, NN * 256);
  gemm(P1, c6W1, c6b1, P3, NN, 256, 128, 1);
  gemm(P3, c6W2, c6b2, (float*)d_out, NN, 128, 64, 0);
}